// PyramidROIAlign_78013785965172
// MI455X (gfx1250) — compile-verified
//
#include <hip/hip_runtime.h>
#include <hip/hip_bf16.h>

// PyramidROIAlign for MI455X (gfx1250, wave32).
// Memory-bound gather (~313MB read, feature maps fit the 192MB L2). Strategy:
//  - 1 workgroup (256 thr / 8 waves) per box; taps+weights computed once into LDS.
//  - Phase 1: horizontal bicubic filter; 256B/wave32 coalesced b64 global loads
//             along the contiguous channel axis; H[28 rows][7 cols][64 ch] -> LDS.
//  - Phase 2: vertical bicubic filter = true matmul out = Wy(7x28) @ H, run as a
//             chain of 7 V_WMMA_F32_16X16X4_F32 (K=28 in chunks of 4). A fragments
//             are built once per box (no predicated LDS loads in the hot loop).

typedef __attribute__((ext_vector_type(2))) float v2f;
typedef __attribute__((ext_vector_type(8))) float v8f;

#define POOLN 7
#define ACOEF (-0.5f)
// H row stride in floats: 7*64 + 16 pad. 464 % 64 == 16, so the two per-chunk
// B loads (rows +0/+1 for lanes 0-15, +2/+3 for lanes 16-31) map to 32 distinct
// LDS banks across the wave -> conflict-free.
#define RSTRIDE 464

__global__ __launch_bounds__(256) void pyramid_roi_align_kernel(
    const float* __restrict__ boxes, const int* __restrict__ ishape,
    const float* __restrict__ p2, const float* __restrict__ p3,
    const float* __restrict__ p4, const float* __restrict__ p5,
    float* __restrict__ out, int nPerBatch, int baseS)
{
  __shared__ float s_wy[28];
  __shared__ float s_wx[28];
  __shared__ int   s_iy[28];
  __shared__ int   s_ix[28];
  __shared__ float s_H[28 * RSTRIDE];   // 51,968 B of the 320 KB WGP LDS

  const int bn    = blockIdx.x;
  const int tid   = threadIdx.x;
  const int batch = bn / nPerBatch;

  const float y1 = boxes[bn * 4 + 0], x1 = boxes[bn * 4 + 1];
  const float y2 = boxes[bn * 4 + 2], x2 = boxes[bn * 4 + 3];

  // roi_level = clip(4 + round(log2(sqrt(h*w) * sqrt(area) / 224)), 2, 5)
  const float area = (float)ishape[0] * (float)ishape[1];
  const float roi  = log2f(sqrtf(fmaxf((y2 - y1) * (x2 - x1), 1e-30f)) *
                           sqrtf(area) * (1.0f / 224.0f));
  const int level  = (int)fminf(fmaxf(4.0f + rintf(roi), 2.0f), 5.0f);
  const int S      = baseS >> (level - 2);
  const float* __restrict__ fm =
      (level == 2) ? p2 : (level == 3) ? p3 : (level == 4) ? p4 : p5;

  // ---- per-box tap indices + cubic weights (14 threads, rest idle) ----
  if (tid < 14) {
    const int axis = tid / POOLN;            // 0 = y, 1 = x
    const int o    = tid - axis * POOLN;     // output coordinate 0..6
    const float Sf = (float)S;
    const float b0 = rintf(y1 * Sf), b1 = rintf(x1 * Sf);
    const float b2 = rintf(y2 * Sf), b3 = rintf(x2 * Sf);
    const float py1 = fminf(b0, b2), py2 = fmaxf(b0, b2);
    const float px1 = fminf(b1, b3), px2 = fmaxf(b1, b3);
    const float validf = ((py1 + px1 + py2 + px2) != 0.0f) ? 1.0f : 0.0f;
    const float lo  = axis ? px1 : py1;
    const float hiV = axis ? px2 : py2;
    const float s = lo + (o + 0.5f) * (1.0f / POOLN) * (hiV - lo) - 0.5f;
    const float f = floorf(s), t = s - f, u = 1.0f - t;
    const float t2 = t * t, t3 = t2 * t, u2 = u * u, u3 = u2 * u;
    float w[4];
    w[0] = ACOEF * (t3 - 2.0f * t2 + t);
    w[1] = (ACOEF + 2.0f) * t3 - (ACOEF + 3.0f) * t2 + 1.0f;
    w[2] = (ACOEF + 2.0f) * u3 - (ACOEF + 3.0f) * u2 + 1.0f;
    w[3] = ACOEF * (t2 - t3);
    const float hic = fmaxf(hiV - 1.0f, lo);
#pragma unroll
    for (int k = 0; k < 4; ++k) {
      float idx = fminf(fmaxf(f - 1.0f + (float)k, lo), hic);
      idx = fminf(fmaxf(idx, 0.0f), (float)(S - 1));
      if (axis) { s_ix[4 * o + k] = (int)idx; s_wx[4 * o + k] = w[k]; }
      else      { s_iy[4 * o + k] = (int)idx; s_wy[4 * o + k] = w[k] * validf; }
    }
  }
  __syncthreads();

  const size_t rowPitch = (size_t)S * 256;
  const float* __restrict__ fmB = fm + (size_t)batch * S * rowPitch;

  const int wave = tid >> 5, lane = tid & 31;
  const int m = lane & 15, hi2 = (lane >> 4) * 2;   // A/B K-halves per ISA layout

  // ---- A fragments for the WMMA chain: loop-invariant, built once per box.
  // A (16x4 f32) layout: VGPR0 lanes0-15 K=0 / lanes16-31 K=2; VGPR1 K=1/K=3.
  // Chunk kc is nonzero only in row m==kc, so D row i=kc gets its 4 y-taps.
  v2f afrag[POOLN];
#pragma unroll
  for (int kc = 0; kc < POOLN; ++kc) {
    const float w0 = s_wy[4 * kc + hi2];       // unconditional -> ds_load_b64
    const float w1 = s_wy[4 * kc + hi2 + 1];
    const bool on = (m == kc);
    v2f a;
    a.x = on ? w0 : 0.0f;                      // v_cndmask, no exec predication
    a.y = on ? w1 : 0.0f;
    afrag[kc] = a;
  }

  const int c2 = lane;          // float2 channel group within 64-ch tile
  // (tid>>5)==wave used as the phase-1 row/col group -> uniform per wave

  for (int ct = 0; ct < 4; ++ct) {                  // 4 channel tiles of 64
    if (ct) __syncthreads();                        // H reuse across tiles
    const int chbase = ct * 64;

    // ---- Phase 1: horizontal cubic filter -> LDS.
    // Each thread filters 2 channels; wave loads 32 lanes x 8B = 256B contiguous.
    {
      const float* __restrict__ fmC = fmB + chbase + 2 * c2;
      for (int rj = wave; rj < 28 * POOLN; rj += 8) {
        const int r = rj / POOLN, j = rj - r * POOLN;
        const float* __restrict__ rowp = fmC + (size_t)s_iy[r] * rowPitch;
        float a0 = 0.0f, a1 = 0.0f;
#pragma unroll
        for (int l = 0; l < 4; ++l) {
          const float wl = s_wx[4 * j + l];
          const float2 v = *(const float2*)&rowp[(size_t)s_ix[4 * j + l] * 256];
          a0 = fmaf(wl, v.x, a0);
          a1 = fmaf(wl, v.y, a1);
        }
        *(float2*)&s_H[r * RSTRIDE + j * 64 + 2 * c2] =
            make_float2(a0, a1);                    // single ds_store_b64
      }
    }
    __syncthreads();

    // ---- Phase 2: vertical cubic filter = Wy(7x28) @ H via WMMA f32 16x16x4
    for (int nt = wave; nt < 28; nt += 8) {         // N-tiles of 16 (j,c) cols
      const int j = nt >> 2, cb = (nt & 3) * 16;
      const int jcidx = j * 64 + cb + m;
      v8f acc = {};
#pragma unroll
      for (int kc = 0; kc < POOLN; ++kc) {
        // B (4x16 f32): lanes0-15 hold K=0(b.x)/K=1(b.y), lanes16-31 K=2/K=3
        const float* hp = &s_H[(4 * kc + hi2) * RSTRIDE + jcidx];
        v2f b;
        b.x = hp[0];
        b.y = hp[RSTRIDE];
        acc = __builtin_amdgcn_wmma_f32_16x16x4_f32(
            false, afrag[kc], false, b, (short)0, acc, false, false);
      }
      // D layout: VGPR v holds row M=v (lanes 0-15) / M=v+8 (lanes 16-31).
      // Only rows 0..6 are real outputs -> lanes 0-15, v<7.
      if (hi2 == 0) {
        float* __restrict__ op =
            out + ((size_t)bn * 49 + j) * 256 + chbase + cb + m;
#pragma unroll
        for (int v = 0; v < POOLN; ++v)
          op[(size_t)v * POOLN * 256] = acc[v];
      }
    }
  }
}

extern "C" void kernel_launch(void* const* d_in, const int* in_sizes, int n_in,
                              void* d_out, int out_size, void* d_ws, size_t ws_size,
                              hipStream_t stream) {
  (void)n_in; (void)d_ws; (void)ws_size; (void)out_size;
  const float* boxes  = (const float*)d_in[0];
  const int*   ishape = (const int*)d_in[1];
  const float* p2     = (const float*)d_in[2];
  const float* p3     = (const float*)d_in[3];
  const float* p4     = (const float*)d_in[4];
  const float* p5     = (const float*)d_in[5];
  float* out          = (float*)d_out;

  const int totalBoxes = in_sizes[0] / 4;               // B*N
  int B = in_sizes[2] / (256 * 256 * 256);              // p2 = B*256*256*C
  if (B < 1) B = 1;
  const int nPer  = totalBoxes / B;
  const int baseS = 256;                                // p2 spatial size

  hipLaunchKernelGGL(pyramid_roi_align_kernel, dim3(totalBoxes), dim3(256), 0,
                     stream, boxes, ishape, p2, p3, p4, p5, out, nPer, baseS);
}